// GraphConvLayer_25598005084524
// MI455X (gfx1250) — compile-verified
//
#include <hip/hip_runtime.h>

typedef __attribute__((ext_vector_type(2))) float v2f;
typedef __attribute__((ext_vector_type(8))) float v8f;

#define GH 256
#define GW 256
#define NF 64
#define NNODES (GH * GW)
#define TILE_X 64
#define AX_STRIDE 68   // floats; 68 mod 64 = 4 -> conflict-free A-frag loads
#define W_STRIDE 72    // floats; 2*72 mod 64 = 16 -> disjoint bank windows for half-waves

__global__ __launch_bounds__(256)
void gcn_fused_wmma(const float* __restrict__ X,
                    const float* __restrict__ Wg,
                    const float* __restrict__ bias,
                    float* __restrict__ out)
{
    __shared__ float AXs[TILE_X * AX_STRIDE];  // 17,408 B
    __shared__ float Ws[NF * W_STRIDE];        // 18,432 B

    const int t = threadIdx.x;
    const int quarter = blockIdx.x;   // 0..3  (64-node chunk within a grid row)
    const int y       = blockIdx.y;   // 0..255 grid row
    const int b       = blockIdx.z;   // batch

    const size_t rowBase = ((size_t)b * NNODES + (size_t)y * GW) * NF; // float idx
    const int x0 = quarter * TILE_X;

    // ---------- Phase 1: 4-neighbor stencil -> LDS ----------
    {
        const int f4   = t & 15;   // float4 chunk within node (0..15)
        const int nsub = t >> 4;   // node sub-index within pass (0..15)
        #pragma unroll
        for (int pass = 0; pass < 4; ++pass) {
            const int nl = pass * 16 + nsub;       // local node 0..63
            const int xg = x0 + nl;                // global x
            const size_t nodeOff = rowBase + (size_t)xg * NF + (size_t)f4 * 4;
            float ax = 0.f, ay = 0.f, az = 0.f, aw = 0.f;
            if (y > 0) {
                const float4 v = *(const float4*)(X + nodeOff - (size_t)GW * NF);
                ax += v.x; ay += v.y; az += v.z; aw += v.w;
            }
            if (y < GH - 1) {
                const float4 v = *(const float4*)(X + nodeOff + (size_t)GW * NF);
                ax += v.x; ay += v.y; az += v.z; aw += v.w;
            }
            if (xg > 0) {
                const float4 v = *(const float4*)(X + nodeOff - NF);
                ax += v.x; ay += v.y; az += v.z; aw += v.w;
            }
            if (xg < GW - 1) {
                const float4 v = *(const float4*)(X + nodeOff + NF);
                ax += v.x; ay += v.y; az += v.z; aw += v.w;
            }
            float4* dst = (float4*)&AXs[nl * AX_STRIDE + f4 * 4]; // 272 B row stride, 16B aligned
            *dst = make_float4(ax, ay, az, aw);
        }
    }

    // ---------- Phase 2: stage W (64x64) into LDS with stride 72 ----------
    {
        const float4* Wv = (const float4*)Wg;  // 1024 float4
        #pragma unroll
        for (int i = 0; i < 4; ++i) {
            const int idx4 = t + 256 * i;      // float4 index
            const int krow = idx4 >> 4;        // f_in row
            const int c4   = idx4 & 15;        // float4 col
            float4* dst = (float4*)&Ws[krow * W_STRIDE + c4 * 4]; // 288 B stride, aligned
            *dst = Wv[idx4];
        }
    }
    __syncthreads();

    // ---------- Phase 3: WMMA  D = AX * W + bias ----------
    const int wave   = t >> 5;
    const int lane   = t & 31;
    const int m_tile = wave & 3;          // which 16-node group
    const int n_base = (wave >> 2) * 32;  // f_out base (0 or 32)
    const int hi     = lane >> 4;         // half-wave select
    const int l15    = lane & 15;

    v8f c0 = {};
    v8f c1 = {};
    const int arow = m_tile * 16 + l15;

    #pragma unroll
    for (int k0 = 0; k0 < NF; k0 += 4) {
        const int kb = k0 + 2 * hi;       // K pair base for this half-wave
        v2f a;
        a.x = AXs[arow * AX_STRIDE + kb];
        a.y = AXs[arow * AX_STRIDE + kb + 1];
        v2f b0, b1;
        b0.x = Ws[kb * W_STRIDE + n_base + l15];
        b0.y = Ws[(kb + 1) * W_STRIDE + n_base + l15];
        b1.x = Ws[kb * W_STRIDE + n_base + 16 + l15];
        b1.y = Ws[(kb + 1) * W_STRIDE + n_base + 16 + l15];
        c0 = __builtin_amdgcn_wmma_f32_16x16x4_f32(false, a, false, b0, (short)0, c0, false, false);
        c1 = __builtin_amdgcn_wmma_f32_16x16x4_f32(false, a, false, b1, (short)0, c1, false, false);
    }

    // ---------- Bias + store ----------
    const float bias0 = bias[n_base + l15];
    const float bias1 = bias[n_base + 16 + l15];
    const size_t outBase = rowBase + (size_t)x0 * NF;
    #pragma unroll
    for (int r = 0; r < 8; ++r) {
        const int node_m = m_tile * 16 + r + 8 * hi;   // D layout: VGPR r -> M=r (lo), r+8 (hi)
        float* o = out + outBase + (size_t)node_m * NF;
        o[n_base + l15]      = c0[r] + bias0;
        o[n_base + 16 + l15] = c1[r] + bias1;
    }
}

extern "C" void kernel_launch(void* const* d_in, const int* in_sizes, int n_in,
                              void* d_out, int out_size, void* d_ws, size_t ws_size,
                              hipStream_t stream) {
    (void)n_in; (void)out_size; (void)d_ws; (void)ws_size;
    const float* X    = (const float*)d_in[0];
    const float* Wg   = (const float*)d_in[1];
    const float* bias = (const float*)d_in[2];
    float* out = (float*)d_out;

    const int nBatch = in_sizes[0] / (NNODES * NF);  // = 8
    dim3 grid(GW / TILE_X, GH, nBatch);
    gcn_fused_wmma<<<grid, 256, 0, stream>>>(X, Wg, bias, out);
}